// GCN_57123065037237
// MI455X (gfx1250) — compile-verified
//
#include <hip/hip_runtime.h>

// CDNA5 / gfx1250. wave32.
typedef __attribute__((ext_vector_type(2))) float v2f;
typedef __attribute__((ext_vector_type(8))) float v8f;

#define N_NODES 100000
#define N_EDGES 1600000
#define IN_F 256
#define OUT_F 128
#define ROW_TILES (N_NODES / 16)        // 6250, exact
#define EDGES_PER_WAVE 64               // N_EDGES % 64 == 0 -> no tail

// ---------------------------------------------------------------------------
// Kernel 1: out[n][f] = bias[f]  (segment_sum of empty segment is 0 + bias)
// ---------------------------------------------------------------------------
__global__ void gcn_bias_init(const float* __restrict__ bias,
                              float* __restrict__ out) {
    int idx = blockIdx.x * blockDim.x + threadIdx.x;
    if (idx < N_NODES * OUT_F) out[idx] = bias[idx & (OUT_F - 1)];
}

// ---------------------------------------------------------------------------
// Kernel 2: support = x @ W via V_WMMA_F32_16X16X4_F32 (full f32 precision).
// One wave computes a 16-row x 128-col strip: 8 accumulator tiles, 64 K-steps.
//
// W is staged in LDS *pair-interleaved*: wlds[p*2*OUT_F + 2n + parity] holds
// W[2p + parity][n]. A B-fragment (two K-adjacent values for one column) is
// then a single 8B-aligned ds_load_b64 straight into an even VGPR pair --
// eliminating the v_dual_mov repacking seen in round 1.
//
// f32 WMMA VGPR layouts (ISA 7.12.2):
//   A 16x4:  lanes 0-15: M=lane, v0=K0,v1=K1 ; lanes 16-31: M=lane-16, v0=K2,v1=K3
//   B 4x16:  lanes 0-15: N=lane, v0=K0,v1=K1 ; lanes 16-31: N=lane-16, v0=K2,v1=K3
//   C/D 16x16: VGPR r: lanes 0-15 -> M=r, N=lane ; lanes 16-31 -> M=r+8, N=lane-16
// ---------------------------------------------------------------------------
__global__ void __launch_bounds__(256)
gcn_gemm_wmma(const float* __restrict__ x,
              const float* __restrict__ w,
              float* __restrict__ support) {
    __shared__ float wlds[IN_F * OUT_F];   // 128 KB staged, pair-interleaved

    // Cooperative stage of W into LDS with pair-interleave transpose.
    // src: w[k*OUT_F + n]  ->  dst: wlds[(k>>1)*(2*OUT_F) + 2n + (k&1)]
    for (int i = threadIdx.x; i < IN_F * OUT_F; i += 256) {
        const int k = i >> 7;              // i / OUT_F
        const int n = i & (OUT_F - 1);
        wlds[(k >> 1) * (2 * OUT_F) + (n << 1) + (k & 1)] = w[i];
    }
    __syncthreads();

    const int lane = threadIdx.x & 31;
    const int wave = threadIdx.x >> 5;
    const int tile = blockIdx.x * 8 + wave;
    if (tile >= ROW_TILES) return;         // whole-wave exit: EXEC all-1s for WMMA

    const int row0 = tile * 16;
    const int half = lane >> 4;            // 0: K pair {0,1}, 1: K pair {2,3}
    const int lr   = lane & 15;

    const float* arow = x + (size_t)(row0 + lr) * IN_F + 2 * half;

    v8f acc[8] = {};                        // 8 N-tiles x 8 VGPRs f32

    for (int k = 0; k < IN_F; k += 4) {
        // A fragment: one b64 load (8B aligned: k%4==0, +2*half*4).
        const v2f a = *(const v2f*)(arow + k);

        // B fragments: pair index p = k/2 + half, column n0 + lr.
        const float* bbase = wlds + ((k >> 1) + half) * (2 * OUT_F) + (lr << 1);
#pragma unroll
        for (int t = 0; t < 8; ++t) {
            const v2f b = *(const v2f*)(bbase + t * 32);   // ds_load_b64
            // 8 args: (neg_a, A, neg_b, B, c_mod, C, reuse_a, reuse_b)
            acc[t] = __builtin_amdgcn_wmma_f32_16x16x4_f32(
                false, a, false, b, (short)0, acc[t], false, false);
        }
    }

    // Store D tiles: VGPR r -> row (r + 8*half), col (n0 + lr).
    float* srow = support + (size_t)row0 * OUT_F + lr;
#pragma unroll
    for (int t = 0; t < 8; ++t) {
#pragma unroll
        for (int r = 0; r < 8; ++r) {
            srow[(size_t)(r + 8 * half) * OUT_F + t * 16] = acc[t][r];
        }
    }
}

// ---------------------------------------------------------------------------
// Kernel 3: out += segment_sum(support[col] * vals, row).
// row is sorted: one wave handles 64 consecutive edges, lane j owns output
// features [4j, 4j+4). Edge metadata is loaded cooperatively (2 edges/lane)
// and broadcast per edge via __shfl (ds_bpermute) instead of 32-lane
// redundant VMEM broadcast loads. Same-row runs accumulate in registers
// (avg degree 16), flushing with native global_atomic_add_f32 only at row
// transitions / chunk ends.
// ---------------------------------------------------------------------------
__device__ __forceinline__ void flush_row(float* __restrict__ out, int r, int f0,
                                          float ax, float ay, float az, float aw) {
    float* o = out + (size_t)r * OUT_F + f0;
    __hip_atomic_fetch_add(o + 0, ax, __ATOMIC_RELAXED, __HIP_MEMORY_SCOPE_AGENT);
    __hip_atomic_fetch_add(o + 1, ay, __ATOMIC_RELAXED, __HIP_MEMORY_SCOPE_AGENT);
    __hip_atomic_fetch_add(o + 2, az, __ATOMIC_RELAXED, __HIP_MEMORY_SCOPE_AGENT);
    __hip_atomic_fetch_add(o + 3, aw, __ATOMIC_RELAXED, __HIP_MEMORY_SCOPE_AGENT);
}

__global__ void __launch_bounds__(256)
gcn_spmm(const float* __restrict__ support,
         const int* __restrict__ row,
         const int* __restrict__ col,
         const float* __restrict__ vals,
         float* __restrict__ out) {
    const int lane = threadIdx.x & 31;
    const int wid  = (blockIdx.x * blockDim.x + threadIdx.x) >> 5;
    const int e0   = wid * EDGES_PER_WAVE;
    if (e0 >= N_EDGES) return;
    const int f0 = lane * 4;

    // Cooperative metadata load: lane l holds edges e0+l and e0+32+l.
    const int   rA = row[e0 + lane];
    const int   rB = row[e0 + 32 + lane];
    const int   cA = col[e0 + lane];
    const int   cB = col[e0 + 32 + lane];
    const float vA = vals[e0 + lane];
    const float vB = vals[e0 + 32 + lane];

    float ax = 0.f, ay = 0.f, az = 0.f, aw = 0.f;
    int cur = -1;

    for (int j = 0; j < EDGES_PER_WAVE; ++j) {
        const int   src = j & 31;                       // uniform
        const int   r = __shfl((j < 32) ? rA : rB, src);
        const int   c = __shfl((j < 32) ? cA : cB, src);
        const float v = __shfl((j < 32) ? vA : vB, src);

        if (r != cur) {                                 // uniform branch
            if (cur >= 0) flush_row(out, cur, f0, ax, ay, az, aw);
            ax = ay = az = aw = 0.f;
            cur = r;
        }
        const float4 g = *(const float4*)(support + (size_t)c * OUT_F + f0);
        ax += g.x * v; ay += g.y * v; az += g.z * v; aw += g.w * v;
    }
    if (cur >= 0) flush_row(out, cur, f0, ax, ay, az, aw);
}

// ---------------------------------------------------------------------------
// Launch. Inputs (setup_inputs order): x, row, col, vals, weight, bias.
// d_ws holds support [N_NODES x OUT_F] f32 = 51.2 MB.
// ---------------------------------------------------------------------------
extern "C" void kernel_launch(void* const* d_in, const int* in_sizes, int n_in,
                              void* d_out, int out_size, void* d_ws, size_t ws_size,
                              hipStream_t stream) {
    const float* x      = (const float*)d_in[0];
    const int*   row    = (const int*)d_in[1];
    const int*   col    = (const int*)d_in[2];
    const float* vals   = (const float*)d_in[3];
    const float* weight = (const float*)d_in[4];
    const float* bias   = (const float*)d_in[5];
    float* out     = (float*)d_out;
    float* support = (float*)d_ws;

    gcn_bias_init<<<(N_NODES * OUT_F + 255) / 256, 256, 0, stream>>>(bias, out);

    gcn_gemm_wmma<<<(ROW_TILES + 7) / 8, 256, 0, stream>>>(x, weight, support);

    const int n_waves = N_EDGES / EDGES_PER_WAVE;       // 25000
    gcn_spmm<<<(n_waves + 7) / 8, 256, 0, stream>>>(support, row, col, vals, out);
}